// GCN_38869454028884
// MI455X (gfx1250) — compile-verified
//
#include <hip/hip_runtime.h>

typedef __attribute__((ext_vector_type(2))) float v2f;
typedef __attribute__((ext_vector_type(4))) float v4f;
typedef __attribute__((ext_vector_type(8))) float v8f;

#define F_IN  128
#define F_HID 64
#define F_OUT 40

// ---------------- degree / normalization ----------------

__global__ void deg_init_kernel(float* __restrict__ deg, int N) {
    int i = blockIdx.x * blockDim.x + threadIdx.x;
    if (i < N) deg[i] = 1.0f;   // self-loop contributes 1 to every node's degree
}

__global__ void deg_edge_kernel(const int* __restrict__ dst, float* __restrict__ deg, int E) {
    int e = blockIdx.x * blockDim.x + threadIdx.x;
    if (e < E) unsafeAtomicAdd(&deg[dst[e]], 1.0f);  // global_atomic_add_f32
}

__global__ void dinv_kernel(const float* __restrict__ deg, float* __restrict__ dinv, int N) {
    int i = blockIdx.x * blockDim.x + threadIdx.x;
    if (i < N) dinv[i] = rsqrtf(deg[i]);
}

// ---------------- weight repack into WMMA-B fragment order ----------------
// Layout: pW[kstep*256 + lane*8 + (tile*2 + t)] = W[(4*kstep + kh + t)*ldw + tile*16 + lm]
// where lm = lane&15, kh = (lane>>4)*2. Lane then loads its 8 floats with two b128s.

__global__ void repack_w1_kernel(const float* __restrict__ W1, float* __restrict__ pW1) {
    int idx = blockIdx.x * blockDim.x + threadIdx.x;   // 32 ksteps * 256 = 8192
    if (idx >= (F_IN / 4) * 256) return;
    int kk   = idx >> 8;
    int rem  = idx & 255;
    int lane = rem >> 3;
    int j    = rem & 7;
    int n    = j >> 1;
    int t    = j & 1;
    int lm   = lane & 15;
    int kh   = (lane >> 4) << 1;
    pW1[idx] = W1[(size_t)(4 * kk + kh + t) * F_HID + n * 16 + lm];
}

__global__ void repack_w2_kernel(const float* __restrict__ W2, float* __restrict__ pW2) {
    int idx = blockIdx.x * blockDim.x + threadIdx.x;   // 16 ksteps * 256 = 4096 (j=6,7 pad)
    if (idx >= (F_HID / 4) * 256) return;
    int kk   = idx >> 8;
    int rem  = idx & 255;
    int lane = rem >> 3;
    int j    = rem & 7;
    int n    = j >> 1;
    int t    = j & 1;
    int lm   = lane & 15;
    int kh   = (lane >> 4) << 1;
    int col  = n * 16 + lm;
    float v  = 0.0f;
    if (j < 6 && col < F_OUT)
        v = W2[(size_t)(4 * kk + kh + t) * F_OUT + col];
    pW2[idx] = v;
}

// ---------------- GEMM1: h1 = x @ W1  (N x 128 @ 128 x 64), f32 WMMA ----------------
// One wave -> 16 rows x 64 cols. Block = 256 threads = 8 waves = 128 rows.
// Row index clamped (not branched) so EXEC stays all-1s for WMMA; stores are guarded.

__global__ void gemm1_wmma_kernel(const float* __restrict__ x, const float* __restrict__ pW1,
                                  float* __restrict__ h1, int N) {
    const int lane = threadIdx.x & 31;
    const int wave = threadIdx.x >> 5;
    const int row0 = blockIdx.x * 128 + wave * 16;
    const int lm   = lane & 15;            // M (and B's N) index within tile
    const int kh   = (lane >> 4) << 1;     // K sub-offset: lanes 0-15 -> 0, 16-31 -> 2
    int r = row0 + lm;
    if (r > N - 1) r = N - 1;              // clamp: garbage rows computed, never stored
    const float* xrow = x + (size_t)r * F_IN + kh;
    const float* pw   = pW1 + (size_t)lane * 8;

    v8f acc0 = {}, acc1 = {}, acc2 = {}, acc3 = {};

    for (int kk = 0; kk < F_IN / 4; ++kk) {
        const v2f A   = *(const v2f*)(xrow + kk * 4);
        const v4f w01 = *(const v4f*)(pw + (size_t)kk * 256);
        const v4f w23 = *(const v4f*)(pw + (size_t)kk * 256 + 4);
        v2f B0; B0[0] = w01[0]; B0[1] = w01[1];
        v2f B1; B1[0] = w01[2]; B1[1] = w01[3];
        v2f B2; B2[0] = w23[0]; B2[1] = w23[1];
        v2f B3; B3[0] = w23[2]; B3[1] = w23[3];
        acc0 = __builtin_amdgcn_wmma_f32_16x16x4_f32(false, A, false, B0, (short)0, acc0, false, false);
        acc1 = __builtin_amdgcn_wmma_f32_16x16x4_f32(false, A, false, B1, (short)0, acc1, false, false);
        acc2 = __builtin_amdgcn_wmma_f32_16x16x4_f32(false, A, false, B2, (short)0, acc2, false, false);
        acc3 = __builtin_amdgcn_wmma_f32_16x16x4_f32(false, A, false, B3, (short)0, acc3, false, false);
    }

    // D layout: VGPR j, lane L: M = j + (L>=16 ? 8 : 0), Ncol = L&15
    const int hi8 = (lane >> 4) << 3;
    for (int j = 0; j < 8; ++j) {
        int rr = row0 + hi8 + j;
        if (rr < N) {
            float* o = h1 + (size_t)rr * F_HID + lm;
            o[0]  = acc0[j];
            o[16] = acc1[j];
            o[32] = acc2[j];
            o[48] = acc3[j];
        }
    }
}

// ---------------- self-loop init: acc[i,:] = h1[i,:] * dinv[i]^2 ----------------

__global__ void acc1_init_kernel(const float* __restrict__ h1, const float* __restrict__ dinv,
                                 float* __restrict__ acc, int total) {
    int idx = blockIdx.x * blockDim.x + threadIdx.x;
    if (idx < total) {
        float di = dinv[idx >> 6];         // node = idx / 64
        acc[idx] = h1[idx] * di * di;
    }
}

// ---------------- edge scatter, layer 1: acc[d] += h1[s] * dinv[s]*dinv[d] ----------------
// One wave per edge; 32 lanes x float2 = 64 features (coalesced 256B row gather).

__global__ void scatter1_kernel(const float* __restrict__ h1, const float* __restrict__ dinv,
                                const int* __restrict__ src, const int* __restrict__ dst,
                                float* __restrict__ acc, int E) {
    const int lane = threadIdx.x & 31;
    const int e = blockIdx.x * (blockDim.x >> 5) + (threadIdx.x >> 5);
    if (e >= E) return;
    const int s = __builtin_amdgcn_readfirstlane(src[e]);   // wave-uniform -> SGPR
    const int d = __builtin_amdgcn_readfirstlane(dst[e]);
    const float norm = dinv[s] * dinv[d];
    const float2 v = *(const float2*)(h1 + (size_t)s * F_HID + lane * 2);
    float* o = acc + (size_t)d * F_HID + lane * 2;
    unsafeAtomicAdd(o,     v.x * norm);
    unsafeAtomicAdd(o + 1, v.y * norm);
}

// ---------------- bias + ReLU: z = max(acc + b1, 0) ----------------

__global__ void relu_bias_kernel(const float* __restrict__ acc, const float* __restrict__ b1,
                                 float* __restrict__ z, int total) {
    int idx = blockIdx.x * blockDim.x + threadIdx.x;
    if (idx < total) {
        float v = acc[idx] + b1[idx & (F_HID - 1)];
        z[idx] = v > 0.0f ? v : 0.0f;
    }
}

// ---------------- GEMM2: h2 = z @ W2  (N x 64 @ 64 x 40), f32 WMMA ----------------
// 3 column tiles (48 wide, zero-padded in packed W2); stores guarded to col < 40.

__global__ void gemm2_wmma_kernel(const float* __restrict__ z, const float* __restrict__ pW2,
                                  float* __restrict__ h2, int N) {
    const int lane = threadIdx.x & 31;
    const int wave = threadIdx.x >> 5;
    const int row0 = blockIdx.x * 128 + wave * 16;
    const int lm   = lane & 15;
    const int kh   = (lane >> 4) << 1;
    int r = row0 + lm;
    if (r > N - 1) r = N - 1;
    const float* zrow = z + (size_t)r * F_HID + kh;
    const float* pw   = pW2 + (size_t)lane * 8;

    v8f acc0 = {}, acc1 = {}, acc2 = {};

    for (int kk = 0; kk < F_HID / 4; ++kk) {
        const v2f A   = *(const v2f*)(zrow + kk * 4);
        const v4f w01 = *(const v4f*)(pw + (size_t)kk * 256);
        const v2f w2v = *(const v2f*)(pw + (size_t)kk * 256 + 4);
        v2f B0; B0[0] = w01[0]; B0[1] = w01[1];
        v2f B1; B1[0] = w01[2]; B1[1] = w01[3];
        v2f B2; B2[0] = w2v[0]; B2[1] = w2v[1];
        acc0 = __builtin_amdgcn_wmma_f32_16x16x4_f32(false, A, false, B0, (short)0, acc0, false, false);
        acc1 = __builtin_amdgcn_wmma_f32_16x16x4_f32(false, A, false, B1, (short)0, acc1, false, false);
        acc2 = __builtin_amdgcn_wmma_f32_16x16x4_f32(false, A, false, B2, (short)0, acc2, false, false);
    }

    const int hi8  = (lane >> 4) << 3;
    const bool c2ok = (lm < 8);            // tile-2 column = 32+lm, valid if < 40
    for (int j = 0; j < 8; ++j) {
        int rr = row0 + hi8 + j;
        if (rr < N) {
            float* o = h2 + (size_t)rr * F_OUT + lm;
            o[0]  = acc0[j];
            o[16] = acc1[j];
            if (c2ok) o[32] = acc2[j];
        }
    }
}

// ---------------- output init: out[i,:] = h2[i,:] * dinv[i]^2 + b2 ----------------

__global__ void out_init_kernel(const float* __restrict__ h2, const float* __restrict__ dinv,
                                const float* __restrict__ b2, float* __restrict__ out, int total) {
    int idx = blockIdx.x * blockDim.x + threadIdx.x;
    if (idx < total) {
        int i = idx / F_OUT;
        int f = idx - i * F_OUT;
        float di = dinv[i];
        out[idx] = h2[idx] * di * di + b2[f];
    }
}

// ---------------- edge scatter, layer 2: out[d] += h2[s] * dinv[s]*dinv[d] ----------------
// One wave per edge; lanes 0-19 carry float2 = 40 features.

__global__ void scatter2_kernel(const float* __restrict__ h2, const float* __restrict__ dinv,
                                const int* __restrict__ src, const int* __restrict__ dst,
                                float* __restrict__ out, int E) {
    const int lane = threadIdx.x & 31;
    const int e = blockIdx.x * (blockDim.x >> 5) + (threadIdx.x >> 5);
    if (e >= E) return;
    const int s = __builtin_amdgcn_readfirstlane(src[e]);
    const int d = __builtin_amdgcn_readfirstlane(dst[e]);
    const float norm = dinv[s] * dinv[d];
    if (lane < 20) {
        const float2 v = *(const float2*)(h2 + (size_t)s * F_OUT + lane * 2);
        float* o = out + (size_t)d * F_OUT + lane * 2;
        unsafeAtomicAdd(o,     v.x * norm);
        unsafeAtomicAdd(o + 1, v.y * norm);
    }
}

// ---------------- launch ----------------

extern "C" void kernel_launch(void* const* d_in, const int* in_sizes, int n_in,
                              void* d_out, int out_size, void* d_ws, size_t ws_size,
                              hipStream_t stream) {
    const float* x  = (const float*)d_in[0];
    const int*   ei = (const int*)d_in[1];
    const float* W1 = (const float*)d_in[2];
    const float* b1 = (const float*)d_in[3];
    const float* W2 = (const float*)d_in[4];
    const float* b2 = (const float*)d_in[5];

    const int N = in_sizes[0] / F_IN;     // 100000
    const int E = in_sizes[1] / 2;        // 1600000
    const int* src = ei;
    const int* dst = ei + E;

    float* ws   = (float*)d_ws;
    float* deg  = ws;                                   // N floats
    float* dinv = ws + (size_t)N;                       // N floats
    float* bufA = ws + (size_t)2 * N;                   // h1, later z   (N*64)
    float* bufB = bufA + (size_t)N * F_HID;             // acc1, later h2 (N*64)
    float* pW1  = bufB + (size_t)N * F_HID;             // 8192 floats
    float* pW2  = pW1 + (size_t)(F_IN / 4) * 256;       // 4096 floats (zero-padded)
    float* out  = (float*)d_out;                        // N*40

    const int B = 256;
    const int nodeBlocks  = (N + B - 1) / B;
    const int edgeBlocks  = (E + B - 1) / B;
    const int hidElems    = N * F_HID;
    const int outElems    = N * F_OUT;
    const int hidBlocks   = (hidElems + B - 1) / B;
    const int outBlocksEl = (outElems + B - 1) / B;
    const int gemmBlocks  = (N + 127) / 128;            // 128 rows / block (8 waves x 16)
    const int waveEdgeBlk = (E + (B / 32) - 1) / (B / 32);
    const int rp1Blocks   = ((F_IN / 4) * 256 + B - 1) / B;
    const int rp2Blocks   = ((F_HID / 4) * 256 + B - 1) / B;

    // one-time weight repack into WMMA fragment order
    repack_w1_kernel<<<rp1Blocks, B, 0, stream>>>(W1, pW1);
    repack_w2_kernel<<<rp2Blocks, B, 0, stream>>>(W2, pW2);

    // degrees + rsqrt normalization
    deg_init_kernel<<<nodeBlocks, B, 0, stream>>>(deg, N);
    deg_edge_kernel<<<edgeBlocks, B, 0, stream>>>(dst, deg, E);
    dinv_kernel<<<nodeBlocks, B, 0, stream>>>(deg, dinv, N);

    // layer 1
    gemm1_wmma_kernel<<<gemmBlocks, B, 0, stream>>>(x, pW1, bufA, N);
    acc1_init_kernel<<<hidBlocks, B, 0, stream>>>(bufA, dinv, bufB, hidElems);
    scatter1_kernel<<<waveEdgeBlk, B, 0, stream>>>(bufA, dinv, src, dst, bufB, E);
    relu_bias_kernel<<<hidBlocks, B, 0, stream>>>(bufB, b1, bufA, hidElems);

    // layer 2
    gemm2_wmma_kernel<<<gemmBlocks, B, 0, stream>>>(bufA, pW2, bufB, N);
    out_init_kernel<<<outBlocksEl, B, 0, stream>>>(bufB, dinv, b2, out, outElems);
    scatter2_kernel<<<waveEdgeBlk, B, 0, stream>>>(bufB, dinv, src, dst, out, E);
}